// WeightedHashEmbedding_8967891714452
// MI455X (gfx1250) — compile-verified
//
#include <hip/hip_runtime.h>

// CDNA5 / gfx1250: wave32, WMMA 16x16x4 f32.
// One wave handles one batch element b:
//   out[b,:] = (1/32) * sum_{c=0..31} weights[idx1[b,c]] * table[idx0[b,c], :]
// K-dimension of the WMMA = chunk groups of 4; B-operand loads ARE the table
// gathers (B's row-striped-across-lanes layout matches a per-lane column load).
// Per-lane ds_bpermute indices fold the lane-half K-selection; 32-bit unsigned
// element offsets let the backend emit SADDR-form global loads (SGPR base +
// 32-bit VGPR offset) instead of 64-bit VGPR address arithmetic.

typedef __attribute__((ext_vector_type(2))) float v2f;
typedef __attribute__((ext_vector_type(8))) float v8f;

#define N_CHUNKS 32
#define DIM      64
#define WAVES_PER_BLOCK 8

__global__ __launch_bounds__(256)
void whe_wmma_kernel(const float* __restrict__ table,
                     const float* __restrict__ weights,
                     const int*   __restrict__ idx0,
                     const int*   __restrict__ idx1,
                     float*       __restrict__ out)
{
    const int lane = threadIdx.x & 31;
    const int wv   = threadIdx.x >> 5;
    const int b    = blockIdx.x * WAVES_PER_BLOCK + wv;   // grid sized exactly: B % 8 == 0

    // Per-lane gather of this batch element's 32 indices + 32 weights.
    // All element offsets fit in u32 (table 16M elems, weights 64M elems).
    const unsigned base = (unsigned)b * N_CHUNKS + lane;
    const int      row  = idx0[base];                      // table row for chunk c = lane
    const float    wgt  = weights[(unsigned)idx1[base]];   // weight for chunk c = lane

    const int half2 = (lane >> 4) << 1;  // 0 for lanes 0-15 (K=0/1), 2 for lanes 16-31 (K=2/3)
    const int col   = lane & 15;         // N index within a 16-dim tile

    // 4 dim-tiles of 16 -> 4 independent 16x16 f32 accumulators (8 VGPRs each).
    v8f acc[4];
    acc[0] = v8f{}; acc[1] = v8f{}; acc[2] = v8f{}; acc[3] = v8f{};

    #pragma unroll
    for (int g = 0; g < 8; ++g) {          // chunk groups: c = 4g .. 4g+3  (K = 0..3)
        const int c = g * 4;

        // A (16x4): A[m][k] = w[c+k] for all m.
        // A-layout: VGPR0 = K0 (lanes 0-15) / K2 (lanes 16-31); VGPR1 = K1 / K3.
        // Per-lane bpermute index selects the right K for each half directly.
        v2f A;
        A.x = __shfl(wgt, c + 0 + half2);
        A.y = __shfl(wgt, c + 1 + half2);

        // B (4x16): B[k][n] = table[row_{c+k}][dtile + n].
        // B-layout: VGPR0 = K0 (lanes 0-15) / K2 (lanes 16-31); VGPR1 = K1 / K3.
        const unsigned rX = (unsigned)__shfl(row, c + 0 + half2);  // K = 0 / 2
        const unsigned rY = (unsigned)__shfl(row, c + 1 + half2);  // K = 1 / 3
        const unsigned oX = rX * DIM + col;    // u32 element offset -> SADDR-form loads
        const unsigned oY = rY * DIM + col;

        #pragma unroll
        for (int t = 0; t < 4; ++t) {       // 4 tiles cover DIM=64; each lane streams
            v2f Bv;                          // 64B-aligned contiguous segments of its rows
            Bv.x = table[oX + t * 16];
            Bv.y = table[oY + t * 16];
            // D = A x B + C   (8 args: neg_a, A, neg_b, B, c_mod, C, reuse_a, reuse_b)
            acc[t] = __builtin_amdgcn_wmma_f32_16x16x4_f32(
                         false, A, false, Bv, (short)0, acc[t], false, false);
        }
    }

    // All 16 rows of each D are identical (A rows identical); VGPR0 of D holds
    // D[0][n] at lanes 0-15. Scale by 1/32 (mean) and store coalesced (64B/tile).
    if (half2 == 0) {
        float* o = out + (unsigned)b * DIM + col;
        #pragma unroll
        for (int t = 0; t < 4; ++t)
            o[t * 16] = acc[t][0] * 0.03125f;
    }
}

extern "C" void kernel_launch(void* const* d_in, const int* in_sizes, int n_in,
                              void* d_out, int out_size, void* d_ws, size_t ws_size,
                              hipStream_t stream) {
    const float* table   = (const float*)d_in[0];   // [ROWS, 64] f32
    const float* weights = (const float*)d_in[1];   // [ROWS*64, 1] f32
    const int*   idx0    = (const int*)  d_in[2];   // [B, 32] i32
    const int*   idx1    = (const int*)  d_in[3];   // [B, 32] i32
    float*       out     = (float*)d_out;           // [B, 64] f32

    const int B = in_sizes[2] / N_CHUNKS;           // 32768
    dim3 grid(B / WAVES_PER_BLOCK);                 // 4096 blocks x 256 threads (8 waves)
    whe_wmma_kernel<<<grid, 256, 0, stream>>>(table, weights, idx0, idx1, out);
}